// BiGeaR_tch_51384988729689
// MI455X (gfx1250) — compile-verified
//
#include <hip/hip_runtime.h>

#define NUM_USERS 200000
#define NUM_ITEMS 60000
#define NNODES    (NUM_USERS + NUM_ITEMS)   // 260000
#define DIM       64
#define NLAYERS   3
#define NTILES    (NNODES / 16)             // 16250, exact (200000 % 16 == 0 too)

typedef __attribute__((ext_vector_type(2))) float v2f;
typedef __attribute__((ext_vector_type(8))) float v8f;

// ---------------------------------------------------------------------------
// Kernel 1: zero layer slots 1..3 of d_out ([N,4,64] layout). Must run every
// call since the scatter kernels accumulate atomically.
// ---------------------------------------------------------------------------
__global__ void bigear_zero_slots(float* __restrict__ out) {
    long i = (long)blockIdx.x * blockDim.x + threadIdx.x;     // one float4 each
    const long total = (long)NNODES * 48;                     // 3 slots * 64 / 4
    if (i >= total) return;
    long n   = i / 48;
    long rem = i - n * 48;
    float4* p = (float4*)(out + n * 256 + 64 + rem * 4);      // skip slot 0
    *p = make_float4(0.f, 0.f, 0.f, 0.f);
}

// ---------------------------------------------------------------------------
// Kernel 2: slot 0 = 0.25 * concat(user_weight, item_weight), computed through
// the matrix pipe: for each 16x16 tile of the embedding,
//   D = sum_{j=0..3} A_j (16x4 slab of X) x B_j (0.25 * shifted identity)
// using V_WMMA_F32_16X16X4_F32.  One wave per 16-node tile (wave-uniform
// early-out keeps EXEC all-ones around every WMMA).
// A 16x4 f32 layout: v0 = {K=0 | K=2}, v1 = {K=1 | K=3} across lane halves.
// B 4x16 f32 layout mirrors A; D 16x16 f32: 8 VGPRs, lane half = M base 0/8.
// ---------------------------------------------------------------------------
__global__ void bigear_init_slot0_wmma(const float* __restrict__ uw,
                                       const float* __restrict__ iw,
                                       float* __restrict__ out) {
    const int lane = threadIdx.x & 31;
    const int wave = threadIdx.x >> 5;
    const int tile = blockIdx.x * (blockDim.x >> 5) + wave;   // uniform per wave
    if (tile >= NTILES) return;                               // whole-wave exit

    const int row0 = tile * 16;
    const float* src = (row0 < NUM_USERS)
                           ? (uw + (long)row0 * DIM)
                           : (iw + (long)(row0 - NUM_USERS) * DIM);

    const int m     = lane & 15;       // matrix row (A) / col (B,D)
    const int hi    = lane >> 4;       // lane half selects K / M-base
    const int kbase = hi * 2;

    // Pre-build the four scaled shifted-identity B_j fragments.
    v2f bfrag[4];
#pragma unroll
    for (int j = 0; j < 4; ++j) {
        bfrag[j].x = (m == 4 * j + (hi ? 2 : 0)) ? 0.25f : 0.0f;
        bfrag[j].y = (m == 4 * j + (hi ? 3 : 1)) ? 0.25f : 0.0f;
    }

#pragma unroll
    for (int ct = 0; ct < 4; ++ct) {                 // four 16-col tiles of DIM
        const int c = ct * 16;
        v8f d = {0.f, 0.f, 0.f, 0.f, 0.f, 0.f, 0.f, 0.f};
#pragma unroll
        for (int j = 0; j < 4; ++j) {
            // A_j: lane (hi,m) holds X[row0+m, c+4j+kbase .. +1]
            v2f a = *(const v2f*)(src + (long)m * DIM + c + 4 * j + kbase);
            d = __builtin_amdgcn_wmma_f32_16x16x4_f32(
                    /*neg_a=*/false, a, /*neg_b=*/false, bfrag[j],
                    /*c_mod=*/(short)0, d, /*reuse_a=*/false, /*reuse_b=*/false);
        }
        // Store D: lane holds rows hi*8..hi*8+7 of column (c+m), slot 0.
        const int mbase = hi * 8;
#pragma unroll
        for (int r = 0; r < 8; ++r) {
            out[(long)(row0 + mbase + r) * 256 + c + m] = d[r];
        }
    }
}

// ---------------------------------------------------------------------------
// Kernel 3: one SpMM layer, scatter-add with HW float atomics.
// Reads slot `layer`, accumulates into slot `layer+1`:
//   out[dst, l+1, :] += (lambda_{l+1}/lambda_l) * val * out[src, l, :]
// 16 lanes per edge, float4 per lane -> one coalesced 256B row gather and
// 4 global_atomic_add_f32 per lane (all L2-resident: table + output < 192MB).
// ---------------------------------------------------------------------------
__global__ void bigear_spmm_layer(const float* __restrict__ eval_,
                                  const int*  __restrict__ esrc,
                                  const int*  __restrict__ edst,
                                  float* __restrict__ out,
                                  int nE, int layer, float ratio) {
    long tid = (long)blockIdx.x * blockDim.x + threadIdx.x;
    long e   = tid >> 4;
    if (e >= nE) return;
    const int l16 = (int)(tid & 15);

    // stream-ahead prefetch of edge arrays (global_prefetch_b8)
    if (e + 2048 < nE) {
        __builtin_prefetch(eval_ + e + 2048, 0, 0);
        __builtin_prefetch(esrc  + e + 2048, 0, 0);
        __builtin_prefetch(edst  + e + 2048, 0, 0);
    }

    const int   s = esrc[e];
    const int   d = edst[e];
    const float w = eval_[e] * ratio;

    const float4 xs = *(const float4*)(out + ((long)s * 4 + layer) * 64 + l16 * 4);
    float* op = out + ((long)d * 4 + layer + 1) * 64 + l16 * 4;

    __hip_atomic_fetch_add(op + 0, w * xs.x, __ATOMIC_RELAXED, __HIP_MEMORY_SCOPE_AGENT);
    __hip_atomic_fetch_add(op + 1, w * xs.y, __ATOMIC_RELAXED, __HIP_MEMORY_SCOPE_AGENT);
    __hip_atomic_fetch_add(op + 2, w * xs.z, __ATOMIC_RELAXED, __HIP_MEMORY_SCOPE_AGENT);
    __hip_atomic_fetch_add(op + 3, w * xs.w, __ATOMIC_RELAXED, __HIP_MEMORY_SCOPE_AGENT);
}

// ---------------------------------------------------------------------------
extern "C" void kernel_launch(void* const* d_in, const int* in_sizes, int n_in,
                              void* d_out, int out_size, void* d_ws, size_t ws_size,
                              hipStream_t stream) {
    const float* uw = (const float*)d_in[0];   // [200000,64]
    const float* iw = (const float*)d_in[1];   // [60000,64]
    const float* ev = (const float*)d_in[2];   // [E]
    const int*   es = (const int*)d_in[3];     // [E]
    const int*   ed = (const int*)d_in[4];     // [E]
    float* out = (float*)d_out;                // [N,4,64]
    const int nE = in_sizes[2];

    // 1) zero accumulation slots 1..3
    {
        long total = (long)NNODES * 48;
        int blocks = (int)((total + 255) / 256);
        bigear_zero_slots<<<blocks, 256, 0, stream>>>(out);
    }
    // 2) slot 0 = lambda0 * x0 via WMMA (one wave per 16-node tile)
    {
        int wavesPerBlock = 256 / 32;
        int blocks = (NTILES + wavesPerBlock - 1) / wavesPerBlock;
        bigear_init_slot0_wmma<<<blocks, 256, 0, stream>>>(uw, iw, out);
    }
    // 3) three SpMM layers; lambda ratio folded into edge weights:
    //    lambdas = {1/4, 2/4, 3/4, 4/4} -> ratios {2, 3/2, 4/3}
    const float ratios[NLAYERS] = {2.0f, 1.5f, 4.0f / 3.0f};
    for (int l = 0; l < NLAYERS; ++l) {
        long threads = (long)nE * 16;
        int blocks = (int)((threads + 255) / 256);
        bigear_spmm_layer<<<blocks, 256, 0, stream>>>(ev, es, ed, out, nE, l, ratios[l]);
    }
}